// PoseEstimationModel_70059506532719
// MI455X (gfx1250) — compile-verified
//
#include <hip/hip_runtime.h>
#include <hip/hip_bf16.h>

// ---------------------------------------------------------------------------
// Pose-estimation depth reprojection loss, MI455X (gfx1250, wave32).
// Bandwidth/scatter bound (~300MB traffic -> ~13us at 23.3TB/s). The 4x4
// homogeneous transform is done with V_WMMA_F32_16X16X4_F32 (full fp32 — the
// int-truncation of projected pixel coords is precision sensitive, so low
// precision WMMA flavors are not appropriate here).
// ---------------------------------------------------------------------------

typedef __attribute__((ext_vector_type(2))) float v2f;
typedef __attribute__((ext_vector_type(8))) float v8f;

#if defined(__gfx1250__) && __has_builtin(__builtin_amdgcn_wmma_f32_16x16x4_f32)
#define USE_WMMA_F32 1
#else
#define USE_WMMA_F32 0
#endif

namespace {
constexpr int H  = 2160;
constexpr int W  = 3840;
constexpr int HW = H * W;
constexpr int TILES = HW / 16;      // 518400 tiles of 16 pixels

// workspace layout (floats)
constexpr int WS_MLAST = 0;    // 16 floats: pose_last @ inv(pose_cur)
constexpr int WS_MCUR  = 16;   // 16 floats: pose_cur  @ inv(pose_cur)
constexpr int WS_REG   = 32;   // reg loss
constexpr int WS_ACC   = 33;   // mse accumulator
constexpr int WS_PROJ  = 64;   // proj_last[HW], proj_cur[HW]
}

// ---- kernel 1: 4x4 inverse + combined matrices + reg loss (1 thread) -------
__global__ void pose_prep_kernel(const float* __restrict__ pose_last,
                                 const float* __restrict__ pose_cur,
                                 float* __restrict__ ws) {
    if (threadIdx.x != 0 || blockIdx.x != 0) return;

    // Gauss-Jordan with partial pivoting: inv(pose_cur)
    float a[4][8];
    for (int r = 0; r < 4; ++r)
        for (int c = 0; c < 4; ++c) {
            a[r][c]     = pose_cur[r * 4 + c];
            a[r][4 + c] = (r == c) ? 1.0f : 0.0f;
        }
    for (int i = 0; i < 4; ++i) {
        int piv = i;
        for (int r = i + 1; r < 4; ++r)
            if (fabsf(a[r][i]) > fabsf(a[piv][i])) piv = r;
        if (piv != i)
            for (int c = 0; c < 8; ++c) { float t = a[i][c]; a[i][c] = a[piv][c]; a[piv][c] = t; }
        float inv = 1.0f / a[i][i];
        for (int c = 0; c < 8; ++c) a[i][c] *= inv;
        for (int r = 0; r < 4; ++r) {
            if (r == i) continue;
            float f = a[r][i];
            for (int c = 0; c < 8; ++c) a[r][c] -= f * a[i][c];
        }
    }
    // Mc = P @ inv(pose_cur)   (row-vector convention: cam = p @ Mc)
    for (int r = 0; r < 4; ++r)
        for (int c = 0; c < 4; ++c) {
            float sL = 0.0f, sC = 0.0f;
            for (int k = 0; k < 4; ++k) {
                sL += pose_last[r * 4 + k] * a[k][4 + c];
                sC += pose_cur [r * 4 + k] * a[k][4 + c];
            }
            ws[WS_MLAST + r * 4 + c] = sL;
            ws[WS_MCUR  + r * 4 + c] = sC;
        }
    float reg = 0.0f;
    for (int i = 0; i < 16; ++i) {
        float d = pose_cur[i] - pose_last[i];
        reg += d * d;
    }
    ws[WS_REG] = reg;
    ws[WS_ACC] = 0.0f;   // zero mse accumulator every call (graph-replay safe)
}

// ---- kernel 2: clear projection buffers (float4 for bandwidth) -------------
__global__ void proj_zero_kernel(float4* __restrict__ pa, float4* __restrict__ pb, int n4) {
    int i = blockIdx.x * blockDim.x + threadIdx.x;
    int stride = gridDim.x * blockDim.x;
    float4 z = {0.0f, 0.0f, 0.0f, 0.0f};
    for (; i < n4; i += stride) { pa[i] = z; pb[i] = z; }
}

// ---- kernel 3: unproject -> WMMA 4x4 transform -> reproject -> scatter -----
// gridDim.y = 0: depth_last with M_last ; 1: depth_current with M_cur
__global__ void project_kernel(const float* __restrict__ depth_last,
                               const float* __restrict__ depth_cur,
                               const float* __restrict__ intr,
                               const float* __restrict__ ws,
                               float* __restrict__ proj_last,
                               float* __restrict__ proj_cur) {
    const int img = blockIdx.y;
    const float* __restrict__ depth = img ? depth_cur  : depth_last;
    const float* __restrict__ Mc    = ws + (img ? WS_MCUR : WS_MLAST);
    float*       __restrict__ proj  = img ? proj_cur   : proj_last;

    const float fx = intr[0], cx = intr[2], fy = intr[4], cy = intr[5];
    const float ifx = 1.0f / fx, ify = 1.0f / fy;

    const int lane = threadIdx.x & 31;
    const int m    = lane & 15;          // point index within 16-pixel tile
    const bool hi  = lane >= 16;
    const int wave   = (blockIdx.x * blockDim.x + threadIdx.x) >> 5;
    const int nwaves = (gridDim.x * blockDim.x) >> 5;

    // A matrix = Mc^T padded to 16x4 (ISA 32-bit A 16x4 layout:
    // VGPR0 = K0 (lanes 0-15) / K2 (lanes 16-31), VGPR1 = K1 / K3)
    v2f amat;
    amat.x = (m < 4) ? Mc[(hi ? 2 : 0) * 4 + m] : 0.0f;
    amat.y = (m < 4) ? Mc[(hi ? 3 : 1) * 4 + m] : 0.0f;

    // uniform-per-wave loop => EXEC is all 1s at the WMMA (required)
    for (int t = wave; t < TILES; t += nwaves) {
        const int p  = t * 16 + m;                 // lanes 16-31 mirror 0-15
        const float Z = depth[p];                  // one 64B line per tile
        const int gy = p / W;
        const int gx = p - gy * W;
        const float X = ((float)gx - cx) * Z * ifx;
        const float Y = ((float)gy - cy) * Z * ify;

        float x, y, z;
#if USE_WMMA_F32
        // B = 4x16 points matrix: VGPR0 = K0 row (X, lanes 0-15) / K2 row
        // (Z, lanes 16-31); VGPR1 = K1 row (Y) / K3 row (1).
        v2f bmat;
        bmat.x = hi ? Z : X;
        bmat.y = hi ? 1.0f : Y;
        v8f c = {};
        c = __builtin_amdgcn_wmma_f32_16x16x4_f32(
                /*neg_a=*/false, amat, /*neg_b=*/false, bmat,
                /*c_mod=*/(short)0, c, /*reuse_a=*/false, /*reuse_b=*/false);
        // D[M=coord, N=point]: lane n<16 holds point n's x/y/z in c[0..2]
        x = c[0]; y = c[1]; z = c[2];
#else
        x = X * Mc[0] + Y * Mc[4] + Z * Mc[8]  + Mc[12];
        y = X * Mc[1] + Y * Mc[5] + Z * Mc[9]  + Mc[13];
        z = X * Mc[2] + Y * Mc[6] + Z * Mc[10] + Mc[14];
#endif
        if (!hi) {
            // trunc-toward-zero matches astype(int32)
            const int u = (int)(x / z * fx + cx);
            const int v = (int)(y / z * fy + cy);
            if (u >= 0 && u < W && v >= 0 && v < H)
                proj[v * W + u] = z;   // racy last-writer-wins == scatter .set
        }
    }
}

// ---- kernel 4: hole-filling combine + squared error reduction --------------
__global__ void loss_kernel(const float* __restrict__ proj_l,
                            const float* __restrict__ proj_c,
                            const float* __restrict__ depth_c,
                            float* __restrict__ acc) {
    __shared__ float s[256];
    float sum = 0.0f;
    int i = blockIdx.x * blockDim.x + threadIdx.x;
    const int stride = gridDim.x * blockDim.x;
    for (; i < HW; i += stride) {
        const float a = proj_l[i];
        const float b = proj_c[i];
        const float mn = fminf(a, b);
        const float mx = fmaxf(a, b);
        const float comb = (mn == 0.0f) ? mx : mn;
        const float d = comb - depth_c[i];
        sum += d * d;
    }
    s[threadIdx.x] = sum;
    __syncthreads();
    for (int off = 128; off > 0; off >>= 1) {
        if ((int)threadIdx.x < off) s[threadIdx.x] += s[threadIdx.x + off];
        __syncthreads();
    }
    if (threadIdx.x == 0) atomicAdd(acc, s[0]);
}

// ---- kernel 5: final scalar -------------------------------------------------
__global__ void finalize_kernel(const float* __restrict__ ws, float* __restrict__ out) {
    if (threadIdx.x == 0 && blockIdx.x == 0)
        out[0] = ws[WS_ACC] / (float)HW + 0.001f * ws[WS_REG];
}

extern "C" void kernel_launch(void* const* d_in, const int* in_sizes, int n_in,
                              void* d_out, int out_size, void* d_ws, size_t ws_size,
                              hipStream_t stream) {
    const float* depth_last = (const float*)d_in[0];
    const float* depth_cur  = (const float*)d_in[1];
    const float* intr       = (const float*)d_in[2];
    const float* pose_last  = (const float*)d_in[3];
    const float* pose_cur   = (const float*)d_in[4];

    float* ws_f      = (float*)d_ws;
    float* proj_last = ws_f + WS_PROJ;
    float* proj_cur  = proj_last + HW;

    pose_prep_kernel<<<1, 32, 0, stream>>>(pose_last, pose_cur, ws_f);
    proj_zero_kernel<<<1024, 256, 0, stream>>>((float4*)proj_last, (float4*)proj_cur, HW / 4);

    dim3 pgrid(512, 2, 1);   // 4096 waves per image slice, ~127 tiles/wave
    project_kernel<<<pgrid, 256, 0, stream>>>(depth_last, depth_cur, intr, ws_f,
                                              proj_last, proj_cur);

    loss_kernel<<<2048, 256, 0, stream>>>(proj_last, proj_cur, depth_cur, ws_f + WS_ACC);
    finalize_kernel<<<1, 32, 0, stream>>>(ws_f, (float*)d_out);
}